// triton_nufft_48704929136774
// MI455X (gfx1250) — compile-verified
//
#include <hip/hip_runtime.h>
#include <math.h>

// ---------------------------------------------------------------------------
// NUFFT forward for MI455X (gfx1250, wave32):
//   1) k_build_w : centered DFT matrix W[k,m] = exp(-2pi i (k-160)(m-160)/320)
//   2) k_pad     : apodize + 1/256 scale + centered zero-pad to 320x320
//   3) k_gemm1   : T = W @ X      (complex GEMM via V_WMMA_F32_16X16X4_F32)
//   4) k_gemm2   : Y = T @ W      (same, output coil-interleaved for gathers)
//   5) k_interp  : 6x6 Kaiser-Bessel gather per point, 8 coils, /36
// ---------------------------------------------------------------------------

typedef __attribute__((ext_vector_type(2))) float v2f;
typedef __attribute__((ext_vector_type(8))) float v8f;

#define NOS   320            // oversampled grid (ceil(1.25*256))
#define NIM   256
#define NC    8
#define NPTS  (1 << 20)
#define GSZ   (NOS * NOS)
#define PADO  32             // (320/2 - 256/2)
#define NTIL  20             // 320/16 tiles per dim

__device__ __constant__ float c_dummy; // keep toolchain happy if unused

constexpr float BETA_F   = 10.9551077f;        // pi*sqrt((4.8*0.75)^2 - 0.8)
constexpr float TWOPI_N  = 0.0196349541f;      // 2*pi/320
constexpr float APOD_K   = 0.0589048623f;      // pi*6/320

// ---------------- device helpers ----------------

__device__ inline float i0f(float x) {
  // Abramowitz & Stegun I0, matching the reference polynomial.
  if (x < 3.75f) {
    float y = x * (1.0f / 3.75f); y *= y;
    return 1.0f + y * (3.5156229f + y * (3.0899424f + y * (1.2067492f +
           y * (0.2659732f + y * (0.0360768f + y * 0.0045813f)))));
  }
  float t = 3.75f / x;
  return (expf(x) * rsqrtf(x)) *
         (0.39894228f + t * (0.01328592f + t * (0.00225319f + t * (-0.00157565f +
          t * (0.00916281f + t * (-0.02057706f + t * (0.02635537f +
          t * (-0.01647633f + t * 0.00392377f))))))));
}

__device__ inline float kbf(float u) {           // u in [-1,1]
  float t = fmaxf(1.0f - u * u, 0.0f);
  return i0f(BETA_F * sqrtf(t));
}

__device__ inline float apod1(int n) {           // n in [0,256)
  float id = (float)(n - NIM / 2);
  float a2 = BETA_F * BETA_F - (APOD_K * id) * (APOD_K * id);
  float a  = sqrtf(a2);
  return a / sinhf(a);
}

__device__ inline v8f cwmma(v2f a, v2f b, v8f c) {
  // D(16x16 f32) = A(16x4) * B(4x16) + C   — full-precision CDNA5 WMMA.
  return __builtin_amdgcn_wmma_f32_16x16x4_f32(
      false, a, false, b, (short)0, c, false, false);
}

// ---------------- kernels ----------------

__global__ void k_build_w(float* __restrict__ wr, float* __restrict__ wi) {
  int idx = blockIdx.x * blockDim.x + threadIdx.x;
  if (idx >= GSZ) return;
  int k = idx / NOS, m = idx % NOS;
  long long a = k - NOS / 2, b = m - NOS / 2;
  int p = (int)((a * b) % NOS);
  if (p < 0) p += NOS;                           // exact phase reduction
  float th = -TWOPI_N * (float)p;
  float s, c;
  __sincosf(th, &s, &c);
  wr[idx] = c;
  wi[idx] = s;
}

__global__ void k_pad(const float* __restrict__ ir, const float* __restrict__ ii,
                      float* __restrict__ xr, float* __restrict__ xi) {
  int idx = blockIdx.x * blockDim.x + threadIdx.x;
  if (idx >= NC * GSZ) return;
  int c = idx / GSZ, r = idx % GSZ;
  int y = r / NOS, x = r % NOS;
  float vr = 0.0f, vi = 0.0f;
  int yy = y - PADO, xx = x - PADO;
  if (yy >= 0 && yy < NIM && xx >= 0 && xx < NIM) {
    float s = apod1(yy) * apod1(xx) * (1.0f / 256.0f);  // apod / sqrt(ny*nx)
    int src = (c * NIM + yy) * NIM + xx;
    vr = ir[src] * s;
    vi = ii[src] * s;
  }
  xr[idx] = vr;
  xi[idx] = vi;
}

// Stage 1: T[c] = W @ X[c].  One wave32 per 16x16 output tile.
__global__ void k_gemm1(const float* __restrict__ wr, const float* __restrict__ wi,
                        const float* __restrict__ xr, const float* __restrict__ xi,
                        float* __restrict__ tr, float* __restrict__ ti) {
  int wave = (blockIdx.x * blockDim.x + threadIdx.x) >> 5;
  int lane = threadIdx.x & 31;
  int c = wave / (NTIL * NTIL);
  int rem = wave % (NTIL * NTIL);
  int tm = rem / NTIL, tn = rem % NTIL;
  int half = lane >> 4, q = lane & 15;
  int m = tm * 16 + q;                 // A row for this lane
  int n = tn * 16 + q;                 // B/C column for this lane
  const float* br_ = xr + (size_t)c * GSZ;
  const float* bi_ = xi + (size_t)c * GSZ;
  v8f accR = {0,0,0,0,0,0,0,0};
  v8f accI = {0,0,0,0,0,0,0,0};
#pragma unroll 2
  for (int k0 = 0; k0 < NOS; k0 += 4) {
    int kA = k0 + half * 2;
    v2f ar  = { wr[m * NOS + kA],  wr[m * NOS + kA + 1] };
    v2f ai  = { wi[m * NOS + kA],  wi[m * NOS + kA + 1] };
    v2f br  = { br_[kA * NOS + n], br_[(kA + 1) * NOS + n] };
    v2f bi  = { bi_[kA * NOS + n], bi_[(kA + 1) * NOS + n] };
    v2f nai = { -ai.x, -ai.y };        // f32 WMMA has no A/B NEG modifier
    accR = cwmma(ar,  br, accR);
    accR = cwmma(nai, bi, accR);
    accI = cwmma(ar,  bi, accI);
    accI = cwmma(ai,  br, accI);
  }
  int row0 = tm * 16 + half * 8;       // C/D layout: VGPR r -> M = r + 8*half
  float* trc = tr + (size_t)c * GSZ;
  float* tic = ti + (size_t)c * GSZ;
#pragma unroll
  for (int r = 0; r < 8; r++) {
    trc[(row0 + r) * NOS + n] = accR[r];
    tic[(row0 + r) * NOS + n] = accI[r];
  }
}

// Stage 2: Y[c] = T[c] @ W, stored coil-interleaved [y][x][c][re,im].
__global__ void k_gemm2(const float* __restrict__ tr, const float* __restrict__ ti,
                        const float* __restrict__ wr, const float* __restrict__ wi,
                        float* __restrict__ yint) {
  int wave = (blockIdx.x * blockDim.x + threadIdx.x) >> 5;
  int lane = threadIdx.x & 31;
  int c = wave / (NTIL * NTIL);
  int rem = wave % (NTIL * NTIL);
  int tm = rem / NTIL, tn = rem % NTIL;
  int half = lane >> 4, q = lane & 15;
  int m = tm * 16 + q;
  int n = tn * 16 + q;
  const float* ar_ = tr + (size_t)c * GSZ;
  const float* ai_ = ti + (size_t)c * GSZ;
  v8f accR = {0,0,0,0,0,0,0,0};
  v8f accI = {0,0,0,0,0,0,0,0};
#pragma unroll 2
  for (int k0 = 0; k0 < NOS; k0 += 4) {
    int kA = k0 + half * 2;
    v2f ar  = { ar_[m * NOS + kA], ar_[m * NOS + kA + 1] };
    v2f ai  = { ai_[m * NOS + kA], ai_[m * NOS + kA + 1] };
    v2f br  = { wr[kA * NOS + n],  wr[(kA + 1) * NOS + n] };
    v2f bi  = { wi[kA * NOS + n],  wi[(kA + 1) * NOS + n] };
    v2f nai = { -ai.x, -ai.y };
    accR = cwmma(ar,  br, accR);
    accR = cwmma(nai, bi, accR);
    accI = cwmma(ar,  bi, accI);
    accI = cwmma(ai,  br, accI);
  }
  int row0 = tm * 16 + half * 8;
#pragma unroll
  for (int r = 0; r < 8; r++) {
    size_t base = ((size_t)(row0 + r) * NOS + n) * (NC * 2) + c * 2;
    yint[base + 0] = accR[r];
    yint[base + 1] = accI[r];
  }
}

// 6x6 KB interpolation; one thread per k-space point, 8 coils in registers.
__global__ void k_interp(const float* __restrict__ trj,
                         const float* __restrict__ yint,
                         float* __restrict__ out) {
  int k = blockIdx.x * blockDim.x + threadIdx.x;
  if (k >= NPTS) return;
  float cy = trj[2 * k + 0] * 1.25f + 160.0f;   // trj * (oy/ny) + oy/2
  float cx = trj[2 * k + 1] * 1.25f + 160.0f;
  int y0 = (int)ceilf(cy - 3.0f);
  int x0 = (int)ceilf(cx - 3.0f);
  float wy[6], wx[6];
  int gy[6], gx[6];
#pragma unroll
  for (int j = 0; j < 6; j++) {
    wy[j] = kbf(((float)(y0 + j) - cy) * (1.0f / 3.0f));
    wx[j] = kbf(((float)(x0 + j) - cx) * (1.0f / 3.0f));
    int a = y0 + j; a += (a < 0) ? NOS : 0; a -= (a >= NOS) ? NOS : 0; gy[j] = a;
    int b = x0 + j; b += (b < 0) ? NOS : 0; b -= (b >= NOS) ? NOS : 0; gx[j] = b;
  }
  float aR[NC] = {0}, aI[NC] = {0};
#pragma unroll
  for (int j = 0; j < 6; j++) {
    const float* rowp = yint + (size_t)gy[j] * (NOS * NC * 2);
#pragma unroll
    for (int l = 0; l < 6; l++) {
      float w = wy[j] * wx[l];
      const float* p = rowp + gx[l] * (NC * 2);   // one 64B contiguous tap
#pragma unroll
      for (int c = 0; c < NC; c++) {
        aR[c] += w * p[2 * c + 0];
        aI[c] += w * p[2 * c + 1];
      }
    }
  }
  const float s = 1.0f / 36.0f;
#pragma unroll
  for (int c = 0; c < NC; c++) {
    size_t o = ((size_t)c * NPTS + (size_t)k) * 2;
    out[o + 0] = aR[c] * s;
    out[o + 1] = aI[c] * s;
  }
}

// ---------------- launcher ----------------

extern "C" void kernel_launch(void* const* d_in, const int* in_sizes, int n_in,
                              void* d_out, int out_size, void* d_ws, size_t ws_size,
                              hipStream_t stream) {
  const float* img_r = (const float*)d_in[0];   // (1,8,256,256)
  const float* img_i = (const float*)d_in[1];   // (1,8,256,256)
  const float* trj   = (const float*)d_in[2];   // (1,2^20,2)

  // Workspace layout (floats): ~20.5 MB total, L2-resident on MI455X.
  float* ws   = (float*)d_ws;
  float* wr   = ws;                  // GSZ
  float* wi   = wr + GSZ;            // GSZ
  float* xr   = wi + GSZ;            // NC*GSZ
  float* xi   = xr + (size_t)NC * GSZ;
  float* tr   = xi + (size_t)NC * GSZ;
  float* ti   = tr + (size_t)NC * GSZ;
  float* yint = ti + (size_t)NC * GSZ;  // 2*NC*GSZ, coil-interleaved grid

  k_build_w<<<(GSZ + 255) / 256, 256, 0, stream>>>(wr, wi);
  k_pad<<<(NC * GSZ + 255) / 256, 256, 0, stream>>>(img_r, img_i, xr, xi);
  // 8 coils * 20*20 tiles = 3200 waves = 400 blocks of 8 wave32s.
  k_gemm1<<<400, 256, 0, stream>>>(wr, wi, xr, xi, tr, ti);
  k_gemm2<<<400, 256, 0, stream>>>(tr, ti, wr, wi, yint);
  k_interp<<<(NPTS + 255) / 256, 256, 0, stream>>>(trj, yint, (float*)d_out);
}